// SelfAttention_Realformer_1906965480093
// MI455X (gfx1250) — compile-verified
//
#include <hip/hip_runtime.h>

// =====================================================================
// Realformer self-attention for MI455X (gfx1250, wave32, WMMA).
//
// Bandwidth bound: preScores read (268MB) + scores write (268MB) + mask
// (~34MB L2-resident) => ~24us at 23.3 TB/s. Flash-style fusion touches the
// [B,H,L,L] tensors exactly once. Softmax bookkeeping is kept under the
// bandwidth roof by (a) accumulating the softmax denominator with a WMMA
// "ones column" (no shuffle sums), (b) a shared running max per 16-row half
// (one shuffle reduce per tile), (c) an LDS float2 LUT for the rel-pos
// scale/bias (no per-element log2).
//
// Workspace (needs >= 16 MB): qbuf/kbuf/vbuf [B][H][L][DK], zbuf [B][L][H][DK].
// Output: z (B*L*C f32) followed by scores (B*H*L*L f32).
// =====================================================================

typedef float v2f __attribute__((ext_vector_type(2)));
typedef float v8f __attribute__((ext_vector_type(8)));

constexpr int B_ = 2, L_ = 2048, C_ = 256, H_ = 8, DK_ = 32;
constexpr int HD_ = H_ * DK_;                       // 256
constexpr float NEGBIG = -4294967295.0f;            // -2^32 + 1
constexpr float INV_SQRT_DK = 0.17677669529663687f; // 1/sqrt(32)

__device__ __forceinline__ v8f wmma4(v2f a, v2f b, v8f c) {
  // D(16x16,f32) = A(16x4,f32) * B(4x16,f32) + C
  return __builtin_amdgcn_wmma_f32_16x16x4_f32(
      /*neg_a=*/false, a, /*neg_b=*/false, b,
      /*c_mod=*/(short)0, c, /*reuse_a=*/false, /*reuse_b=*/false);
}

// ---------------------------------------------------------------------
// QKV projection: X[4096x256] @ W[256x256] + bias, scatter-written as
// [b][h][l][d] so attention sees contiguous K/V per (b,h).
// grid = (256, 1, 3); block = 256 (8 waves, each wave 2 output tiles).
// ---------------------------------------------------------------------
__global__ void qkv_gemm_kernel(const float* __restrict__ qx,
                                const float* __restrict__ kx,
                                const float* __restrict__ vx,
                                const float* __restrict__ wq,
                                const float* __restrict__ wk,
                                const float* __restrict__ wv,
                                const float* __restrict__ bq,
                                const float* __restrict__ bk,
                                const float* __restrict__ bv,
                                float* __restrict__ ws) {
  const int which = blockIdx.z;
  const float* X    = which == 0 ? qx : which == 1 ? kx : vx;
  const float* W    = which == 0 ? wq : which == 1 ? wk : wv;
  const float* bias = which == 0 ? bq : which == 1 ? bk : bv;
  float* Obuf = ws + (size_t)which * ((size_t)B_ * H_ * L_ * DK_);

  const int wave = threadIdx.x >> 5;
  const int lane = threadIdx.x & 31;
  const int laneLo = lane & 15, laneHi = lane >> 4;
  const int mrow = blockIdx.x * 16 + laneLo;

  v8f acc0 = {}; v8f acc1 = {};
  const int n0 = wave * 16;        // tiles n = wave and wave+8
  const int n1 = (wave + 8) * 16;

  for (int k0 = 0; k0 < C_; k0 += 4) {
    v2f a, b0, b1;
    // A 16x4 f32 frag: lanes0-15 hold k0/k0+1, lanes16-31 hold k0+2/k0+3
    const int ka = k0 + 2 * laneHi;
    a.x = X[(size_t)mrow * C_ + ka];
    a.y = X[(size_t)mrow * C_ + ka + 1];
    // B 4x16 f32 frag: v0 rows k0 (lanes0-15) / k0+1 (lanes16-31); v1 +2
    const int kb = k0 + laneHi;
    b0.x = W[(size_t)kb * HD_ + n0 + laneLo];
    b0.y = W[(size_t)(kb + 2) * HD_ + n0 + laneLo];
    b1.x = W[(size_t)kb * HD_ + n1 + laneLo];
    b1.y = W[(size_t)(kb + 2) * HD_ + n1 + laneLo];
    acc0 = wmma4(a, b0, acc0);
    acc1 = wmma4(a, b1, acc1);
  }

#pragma unroll
  for (int e = 0; e < 8; ++e) {
    const int row = blockIdx.x * 16 + e + 8 * laneHi; // C/D: M = e (+8 hi half)
    const int bb = row / L_, l = row % L_;
    {
      const int col = n0 + laneLo, h = col >> 5, d = col & 31;
      Obuf[(((size_t)bb * H_ + h) * L_ + l) * DK_ + d] = acc0[e] + bias[col];
    }
    {
      const int col = n1 + laneLo, h = col >> 5, d = col & 31;
      Obuf[(((size_t)bb * H_ + h) * L_ + l) * DK_ + d] = acc1[e] + bias[col];
    }
  }
}

// ---------------------------------------------------------------------
// Fused attention: one block per (b, h, 16-row block). 8 waves, each owns a
// 256-column strip (16 tiles of 16). Per tile: WMMA QK^T -> LUT scale/bias +
// preScores + mask -> scores out -> online softmax (shared-half max, WMMA
// ones-column denominator) -> WMMA P@V. Cross-wave flash merge at the end.
// ---------------------------------------------------------------------
__global__ void attn_kernel(const float* __restrict__ Qb,
                            const float* __restrict__ Kb,
                            const float* __restrict__ Vb,
                            const float* __restrict__ preS,
                            const int* __restrict__ maskPAD,
                            const float* __restrict__ embK,
                            const float* __restrict__ embB,
                            float* __restrict__ scoresOut,
                            float* __restrict__ Zws) {
  __shared__ float  Qs[16][DK_];         // 2 KB
  __shared__ float2 relLUT[L_];          // 16 KB: {embK[idx][h]/sqrt(DK), embB[idx][h]}
  __shared__ float  Plds[8][16][17];     // per-wave P staging, padded
  __shared__ float  mW[8][16];
  __shared__ float  lW[8][16];
  __shared__ float  Oslab[8][16][DK_];   // 16 KB

  const int rb = blockIdx.x;             // 0..127
  const int h  = blockIdx.y;
  const int b  = blockIdx.z;
  const int wave = threadIdx.x >> 5;
  const int lane = threadIdx.x & 31;
  const int laneLo = lane & 15, laneHi = lane >> 4;
  const int rowbase = rb * 16;

  const size_t bh = (size_t)b * H_ + h;
  const float* Q = Qb + (bh * L_ + rowbase) * DK_;
  const float* K = Kb + bh * L_ * DK_;
  const float* V = Vb + bh * L_ * DK_;
  const float* pre = preS + bh * L_ * L_;
  const int* mp = maskPAD + (size_t)b * L_ * L_;
  float* sOut = scoresOut + bh * L_ * L_;

  for (int t = threadIdx.x; t < 16 * DK_; t += blockDim.x)
    Qs[t / DK_][t % DK_] = Q[t];
  // rel-pos bucket LUT: d = |i-j|, log2 buckets beyond 7, clip [0,14]
  for (int dlt = threadIdx.x; dlt < L_; dlt += blockDim.x) {
    float fd = (float)dlt;
    if (fd > 7.0f) fd = 7.0f + __log2f(fmaxf(fd - 7.0f, 1.0f));
    fd = fminf(fd, 14.0f);
    const int idx = (int)fd;
    relLUT[dlt] = make_float2(embK[idx * H_ + h] * INV_SQRT_DK,
                              embB[idx * H_ + h]);
  }
  __syncthreads();

  float mrun = -3.0e38f;                 // running max, shared per 16-lane half
  v8f oa = {}; v8f ob = {};              // O accumulators (N = 0..15, 16..31)
  v8f osum = {};                         // softmax denominator (ones column)
  v2f bone;                              // B frag: B[k][n] = (n==0)
  bone.x = (laneLo == 0) ? 1.0f : 0.0f;
  bone.y = bone.x;

  for (int t = 0; t < 16; ++t) {
    const int c0 = wave * 256 + t * 16;  // column tile base

    // ---- S tile = Q * K^T (K-dim = 32 -> 8 WMMA steps) ----
    v8f sacc = {};
#pragma unroll
    for (int k0 = 0; k0 < DK_; k0 += 4) {
      v2f a, bf;
      a.x = Qs[laneLo][k0 + 2 * laneHi];
      a.y = Qs[laneLo][k0 + 2 * laneHi + 1];
      bf.x = K[(size_t)(c0 + laneLo) * DK_ + k0 + laneHi];      // K^T[k][n]
      bf.y = K[(size_t)(c0 + laneLo) * DK_ + k0 + 2 + laneHi];
      sacc = wmma4(a, bf, sacc);
    }

    // ---- scores: s = S*kb' + bb + pre, mask, emit; track half-tile max ----
    const int j = c0 + laneLo;
    float tmax = -3.0e38f;
#pragma unroll
    for (int e = 0; e < 8; ++e) {
      const int i = rowbase + e + 8 * laneHi;
      const int delta = (i > j) ? (i - j) : (j - i);
      const float2 rl = relLUT[delta];
      float s = fmaf(sacc[e], rl.x, rl.y) + pre[(size_t)i * L_ + j];
      s = (mp[(size_t)i * L_ + j] == 0) ? NEGBIG : s;
      sOut[(size_t)i * L_ + j] = s;      // scores output (pre-softmax)
      sacc[e] = s;
      tmax = fmaxf(tmax, s);
    }
    // one shuffle reduce per tile (xor masks <16 stay within the half)
    tmax = fmaxf(tmax, __shfl_xor(tmax, 1));
    tmax = fmaxf(tmax, __shfl_xor(tmax, 2));
    tmax = fmaxf(tmax, __shfl_xor(tmax, 4));
    tmax = fmaxf(tmax, __shfl_xor(tmax, 8));

    const float mn = fmaxf(mrun, tmax);
    const float sc = __expf(mrun - mn);
    mrun = mn;
    oa = oa * sc; ob = ob * sc; osum = osum * sc;

#pragma unroll
    for (int e = 0; e < 8; ++e) {
      const float p = __expf(sacc[e] - mn);
      Plds[wave][e + 8 * laneHi][laneLo] = p;   // stage P for A-frag reload
    }
    // cross-lane LDS RAW within the wave: wait out the ds_stores
    asm volatile("s_wait_dscnt 0" ::: "memory");

    // ---- O += P*V; denominator += P*ones (K=16 -> 4 steps, 3 accums) ----
#pragma unroll
    for (int k0 = 0; k0 < 16; k0 += 4) {
      v2f a, b0, b1;
      a.x = Plds[wave][laneLo][k0 + 2 * laneHi];
      a.y = Plds[wave][laneLo][k0 + 2 * laneHi + 1];
      const float* v0 = V + (size_t)(c0 + k0 + laneHi) * DK_;
      const float* v1 = V + (size_t)(c0 + k0 + 2 + laneHi) * DK_;
      b0.x = v0[laneLo];      b0.y = v1[laneLo];
      b1.x = v0[16 + laneLo]; b1.y = v1[16 + laneLo];
      oa = wmma4(a, b0, oa);
      ob = wmma4(a, b1, ob);
      osum = wmma4(a, bone, osum);
    }
  }

  // ---- cross-wave flash merge (l lives in osum column 0: lanes 0/16) ----
  if (laneLo == 0) {
#pragma unroll
    for (int e = 0; e < 8; ++e) {
      mW[wave][e + 8 * laneHi] = mrun;
      lW[wave][e + 8 * laneHi] = osum[e];
    }
  }
  __syncthreads();

#pragma unroll
  for (int e = 0; e < 8; ++e) {
    const int row = e + 8 * laneHi;
    float mg = -3.0e38f;
#pragma unroll
    for (int ww = 0; ww < 8; ++ww) mg = fmaxf(mg, mW[ww][row]);
    const float f = __expf(mrun - mg);
    Oslab[wave][row][laneLo] = oa[e] * f;
    Oslab[wave][row][16 + laneLo] = ob[e] * f;
  }
  __syncthreads();

  for (int t = threadIdx.x; t < 16 * DK_; t += blockDim.x) {
    const int row = t / DK_, d = t % DK_;
    float mg = -3.0e38f;
#pragma unroll
    for (int ww = 0; ww < 8; ++ww) mg = fmaxf(mg, mW[ww][row]);
    float lg = 0.0f;
#pragma unroll
    for (int ww = 0; ww < 8; ++ww) lg += lW[ww][row] * __expf(mW[ww][row] - mg);
    float z = 0.0f;
#pragma unroll
    for (int ww = 0; ww < 8; ++ww) z += Oslab[ww][row][d];
    z /= fmaxf(lg, 1e-37f);
    // Z as [b][l][h][d] (== row-major [B*L][H*DK] for the output GEMM)
    Zws[(((size_t)b * L_ + rowbase + row) * H_ + h) * DK_ + d] = z;
  }
}

// ---------------------------------------------------------------------
// Output projection: Z[4096x256] @ WO[256x256] + bias -> z (row-major).
// ---------------------------------------------------------------------
__global__ void out_gemm_kernel(const float* __restrict__ Z,
                                const float* __restrict__ WO,
                                const float* __restrict__ bo,
                                float* __restrict__ zout) {
  const int wave = threadIdx.x >> 5;
  const int lane = threadIdx.x & 31;
  const int laneLo = lane & 15, laneHi = lane >> 4;
  const int mrow = blockIdx.x * 16 + laneLo;

  v8f acc0 = {}; v8f acc1 = {};
  const int n0 = wave * 16, n1 = (wave + 8) * 16;

  for (int k0 = 0; k0 < HD_; k0 += 4) {
    v2f a, b0, b1;
    const int ka = k0 + 2 * laneHi;
    a.x = Z[(size_t)mrow * HD_ + ka];
    a.y = Z[(size_t)mrow * HD_ + ka + 1];
    const int kb = k0 + laneHi;
    b0.x = WO[(size_t)kb * C_ + n0 + laneLo];
    b0.y = WO[(size_t)(kb + 2) * C_ + n0 + laneLo];
    b1.x = WO[(size_t)kb * C_ + n1 + laneLo];
    b1.y = WO[(size_t)(kb + 2) * C_ + n1 + laneLo];
    acc0 = wmma4(a, b0, acc0);
    acc1 = wmma4(a, b1, acc1);
  }

#pragma unroll
  for (int e = 0; e < 8; ++e) {
    const int row = blockIdx.x * 16 + e + 8 * laneHi;
    zout[(size_t)row * C_ + n0 + laneLo] = acc0[e] + bo[n0 + laneLo];
    zout[(size_t)row * C_ + n1 + laneLo] = acc1[e] + bo[n1 + laneLo];
  }
}

// ---------------------------------------------------------------------
extern "C" void kernel_launch(void* const* d_in, const int* in_sizes, int n_in,
                              void* d_out, int out_size, void* d_ws, size_t ws_size,
                              hipStream_t stream) {
  const float* qx   = (const float*)d_in[0];
  const float* kx   = (const float*)d_in[1];
  const float* vx   = (const float*)d_in[2];
  const float* preS = (const float*)d_in[3];
  const int*   mask = (const int*)d_in[4];
  const float* WQ_w = (const float*)d_in[5];
  const float* WQ_b = (const float*)d_in[6];
  const float* WK_w = (const float*)d_in[7];
  const float* WK_b = (const float*)d_in[8];
  const float* WV_w = (const float*)d_in[9];
  const float* WV_b = (const float*)d_in[10];
  const float* WO_w = (const float*)d_in[11];
  const float* WO_b = (const float*)d_in[12];
  const float* embK = (const float*)d_in[13];
  const float* embB = (const float*)d_in[14];
  (void)in_sizes; (void)n_in; (void)out_size; (void)ws_size;

  float* ws = (float*)d_ws;
  const size_t SZ = (size_t)B_ * H_ * L_ * DK_;   // 1,048,576 f32 each
  float* qbuf = ws;
  float* kbuf = ws + SZ;
  float* vbuf = ws + 2 * SZ;
  float* zbuf = ws + 3 * SZ;

  float* zOut = (float*)d_out;
  float* scoresOut = zOut + (size_t)B_ * L_ * C_;

  qkv_gemm_kernel<<<dim3(B_ * L_ / 16, 1, 3), 256, 0, stream>>>(
      qx, kx, vx, WQ_w, WK_w, WV_w, WQ_b, WK_b, WV_b, ws);

  attn_kernel<<<dim3(L_ / 16, H_, B_), 256, 0, stream>>>(
      qbuf, kbuf, vbuf, preS, mask, embK, embB, scoresOut, zbuf);

  out_gemm_kernel<<<dim3(B_ * L_ / 16, 1, 1), 256, 0, stream>>>(
      zbuf, WO_w, WO_b, zOut);
}